// dense_baens_21844203667565
// MI455X (gfx1250) — compile-verified
//
#include <hip/hip_runtime.h>

typedef float v2f __attribute__((ext_vector_type(2)));
typedef float v8f __attribute__((ext_vector_type(8)));

constexpr int kN  = 2048;
constexpr int kD1 = 512;
constexpr int kD2 = 512;
constexpr int kColsPerTile  = 16;            // WMMA N dimension
constexpr int kTilesPerRow  = kD2 / kColsPerTile;   // 32
constexpr int kWavesPerBlk  = 8;             // 256 threads, wave32

// One wave32 computes out[n, e0:e0+16] = x[n,:] · U[n,:,e0:e0+16] + bias[n,e0:e0+16]
// via 128 chained V_WMMA_F32_16X16X4_F32 (K=4 per step).
//
// A (16x4 f32, 2 VGPRs): lane layout — lanes 0-15 hold K=0 (v0) / K=1 (v1),
// lanes 16-31 hold K=2 / K=3. We broadcast the x chunk to every M row, so all
// 16 rows of D carry the same GEMV result; we store row 0 (C/D VGPR0, lanes 0-15).
//
// B (4x16 f32, 2 VGPRs): lane l supplies column e0+(l&15) of rows
// d + 2*(l>>4) (v0) and d + 2*(l>>4) + 1 (v1). Each 16-lane half reads a
// contiguous 64B row segment -> fully coalesced b32 loads.
__global__ __launch_bounds__(256) void baens_gemv_wmma(
    const float* __restrict__ x,
    const float* __restrict__ U,
    const float* __restrict__ bias,
    float* __restrict__ out)
{
    const int lane = threadIdx.x & 31;
    const int wave = threadIdx.x >> 5;
    const int w    = blockIdx.x * kWavesPerBlk + wave;   // global wave id
    const int n    = w / kTilesPerRow;                   // sample
    const int e0   = (w % kTilesPerRow) * kColsPerTile;  // column-tile base

    const int half = lane >> 4;   // 0 -> K=0/1 rows, 1 -> K=2/3 rows
    const int col  = lane & 15;

    // Accumulator: seed C row 0 (VGPR0, lanes 0-15 = M=0, N=col) with bias.
    v8f c = {};
    if (lane < 16) c[0] = bias[(size_t)n * kD2 + e0 + col];

    const float* xr = x + (size_t)n * kD1 + 2 * half;
    const float* Ur = U + (size_t)n * kD1 * kD2
                        + (size_t)(2 * half) * kD2 + e0 + col;

    #pragma unroll 4
    for (int d = 0; d < kD1; d += 4) {
        // A: x[d + 2*half], x[d + 2*half + 1] broadcast to all 16 M rows.
        v2f a = *(const v2f*)(xr + d);
        // B: 4x16 fp32 tile of U (rows d..d+3, cols e0..e0+15).
        v2f b;
        b.x = Ur[0];          // row d + 2*half
        b.y = Ur[kD2];        // row d + 2*half + 1
        __builtin_prefetch(Ur + 8 * kD2, 0, 1);   // stream U ~16KB (8 rows) ahead
        Ur += 4 * (size_t)kD2;

        c = __builtin_amdgcn_wmma_f32_16x16x4_f32(
                /*neg_a=*/false, a, /*neg_b=*/false, b,
                /*c_mod=*/(short)0, c, /*reuse_a=*/false, /*reuse_b=*/false);
    }

    if (lane < 16) out[(size_t)n * kD2 + e0 + col] = c[0];
}

extern "C" void kernel_launch(void* const* d_in, const int* in_sizes, int n_in,
                              void* d_out, int out_size, void* d_ws, size_t ws_size,
                              hipStream_t stream)
{
    const float* x    = (const float*)d_in[0];   // (N, D1)
    const float* U    = (const float*)d_in[1];   // (N, D1, D2)
    const float* bias = (const float*)d_in[2];   // (N, 1, D2)
    float*       out  = (float*)d_out;           // (N, D2)

    const int totalWaves = kN * kTilesPerRow;            // 65536
    const int blocks     = totalWaves / kWavesPerBlk;    // 8192

    baens_gemv_wmma<<<blocks, 256, 0, stream>>>(x, U, bias, out);
}